// ImprovedUnitaryDilationNetwork_53807350284466
// MI455X (gfx1250) — compile-verified
//
#include <hip/hip_runtime.h>
#include <hip/hip_bf16.h>
#include <cstddef>

#define NN   1024
#define NEDG 16384
#define HD   64

typedef float v2f __attribute__((ext_vector_type(2)));
typedef float v8f __attribute__((ext_vector_type(8)));

// ---------------------------------------------------------------------------
// Complex GEMM: C = alpha * opA(A) * opB(B) + beta * I
// opA/opB: 0 = N, 1 = conjugate-transpose.
// fp32 path on CDNA5 -> V_WMMA_F32_16X16X4_F32, wave32, 8 waves/block,
// 64x64 C tile per block, each wave owns two 16x16 subtiles (re+im).
// ---------------------------------------------------------------------------
__global__ __launch_bounds__(256) void k_cgemm(
    const float* __restrict__ Ar, const float* __restrict__ Ai,
    const float* __restrict__ Br, const float* __restrict__ Bi,
    float* __restrict__ Cr, float* __restrict__ Ci,
    int n, int opA, int opB, float alpha, float beta)
{
    __shared__ float As_r[64][17];   // [m][k], pad 17 -> conflict-free frag reads
    __shared__ float As_i[64][17];
    __shared__ float Bs_r[16][65];   // [k][nn]
    __shared__ float Bs_i[16][65];

    const int tid  = threadIdx.x;
    const int lane = tid & 31;
    const int w    = tid >> 5;       // wave 0..7
    const int wm   = w >> 2;         // 0..1  (M direction, 32 rows each)
    const int wn   = w & 3;          // 0..3  (N direction, 16 cols each)
    const int half = lane >> 4;      // 0/1 lane half
    const int lr   = lane & 15;
    const int m0   = blockIdx.y * 64;
    const int n0   = blockIdx.x * 64;

    v8f cr0 = {}, ci0 = {}, cr1 = {}, ci1 = {};

    for (int k0 = 0; k0 < n; k0 += 16) {
        // ---- stage A tile (64 x 16, re+im) ----
        {
            const int li = tid * 4;
            if (opA == 0) {
                const int m = li >> 4, k = li & 15;
                const float4 r4 = *(const float4*)(Ar + (size_t)(m0 + m) * n + k0 + k);
                const float4 i4 = *(const float4*)(Ai + (size_t)(m0 + m) * n + k0 + k);
                As_r[m][k + 0] = r4.x; As_r[m][k + 1] = r4.y;
                As_r[m][k + 2] = r4.z; As_r[m][k + 3] = r4.w;
                As_i[m][k + 0] = i4.x; As_i[m][k + 1] = i4.y;
                As_i[m][k + 2] = i4.z; As_i[m][k + 3] = i4.w;
            } else { // conj-transpose: opA(A)[m][k] = conj(A[k][m])
                const int k = li >> 6, m = li & 63;
                const float4 r4 = *(const float4*)(Ar + (size_t)(k0 + k) * n + m0 + m);
                const float4 i4 = *(const float4*)(Ai + (size_t)(k0 + k) * n + m0 + m);
                As_r[m + 0][k] = r4.x; As_r[m + 1][k] = r4.y;
                As_r[m + 2][k] = r4.z; As_r[m + 3][k] = r4.w;
                As_i[m + 0][k] = -i4.x; As_i[m + 1][k] = -i4.y;
                As_i[m + 2][k] = -i4.z; As_i[m + 3][k] = -i4.w;
            }
        }
        // ---- stage B tile (16 x 64, re+im) ----
        {
            const int li = tid * 4;
            if (opB == 0) {
                const int k = li >> 6, nn = li & 63;
                const float4 r4 = *(const float4*)(Br + (size_t)(k0 + k) * n + n0 + nn);
                const float4 i4 = *(const float4*)(Bi + (size_t)(k0 + k) * n + n0 + nn);
                Bs_r[k][nn + 0] = r4.x; Bs_r[k][nn + 1] = r4.y;
                Bs_r[k][nn + 2] = r4.z; Bs_r[k][nn + 3] = r4.w;
                Bs_i[k][nn + 0] = i4.x; Bs_i[k][nn + 1] = i4.y;
                Bs_i[k][nn + 2] = i4.z; Bs_i[k][nn + 3] = i4.w;
            } else { // conj-transpose: opB(B)[k][nn] = conj(B[nn][k])
                const int nn = li >> 4, k = li & 15;
                const float4 r4 = *(const float4*)(Br + (size_t)(n0 + nn) * n + k0 + k);
                const float4 i4 = *(const float4*)(Bi + (size_t)(n0 + nn) * n + k0 + k);
                Bs_r[k + 0][nn] = r4.x; Bs_r[k + 1][nn] = r4.y;
                Bs_r[k + 2][nn] = r4.z; Bs_r[k + 3][nn] = r4.w;
                Bs_i[k + 0][nn] = -i4.x; Bs_i[k + 1][nn] = -i4.y;
                Bs_i[k + 2][nn] = -i4.z; Bs_i[k + 3][nn] = -i4.w;
            }
        }
        __syncthreads();

        if (k0 + 16 < n) {  // hint: next A stage -> global_prefetch_b8
            __builtin_prefetch(Ar + (size_t)(m0 + (tid & 63)) * n + k0 + 16, 0, 1);
        }

        #pragma unroll
        for (int kk = 0; kk < 4; ++kk) {
            const int kb  = kk * 4 + 2 * half;       // f32 A/B frag: K = 2*half + {0,1}
            const int mA0 = wm * 32 + lr;
            const int mA1 = mA0 + 16;
            const int nB  = wn * 16 + lr;

            v2f ar0 = { As_r[mA0][kb], As_r[mA0][kb + 1] };
            v2f ai0 = { As_i[mA0][kb], As_i[mA0][kb + 1] };
            v2f ar1 = { As_r[mA1][kb], As_r[mA1][kb + 1] };
            v2f ai1 = { As_i[mA1][kb], As_i[mA1][kb + 1] };
            v2f br  = { Bs_r[kb][nB],  Bs_r[kb + 1][nB]  };
            v2f bi  = { Bs_i[kb][nB],  Bs_i[kb + 1][nB]  };
            v2f an0 = -ai0, an1 = -ai1;  // f32 WMMA has no A/B NEG bits -> VALU negate

            // Cr += Ar*Br - Ai*Bi ; Ci += Ar*Bi + Ai*Br
            cr0 = __builtin_amdgcn_wmma_f32_16x16x4_f32(false, ar0, false, br, (short)0, cr0, false, false);
            cr0 = __builtin_amdgcn_wmma_f32_16x16x4_f32(false, an0, false, bi, (short)0, cr0, false, false);
            ci0 = __builtin_amdgcn_wmma_f32_16x16x4_f32(false, ar0, false, bi, (short)0, ci0, false, false);
            ci0 = __builtin_amdgcn_wmma_f32_16x16x4_f32(false, ai0, false, br, (short)0, ci0, false, false);

            cr1 = __builtin_amdgcn_wmma_f32_16x16x4_f32(false, ar1, false, br, (short)0, cr1, false, false);
            cr1 = __builtin_amdgcn_wmma_f32_16x16x4_f32(false, an1, false, bi, (short)0, cr1, false, false);
            ci1 = __builtin_amdgcn_wmma_f32_16x16x4_f32(false, ar1, false, bi, (short)0, ci1, false, false);
            ci1 = __builtin_amdgcn_wmma_f32_16x16x4_f32(false, ai1, false, br, (short)0, ci1, false, false);
        }
        __syncthreads();
    }

    // epilogue: C layout VGPR r, lane l -> M = r + 8*(l>>4), N = l&15
    const int colg = n0 + wn * 16 + lr;
    #pragma unroll
    for (int r = 0; r < 8; ++r) {
        const int row0 = m0 + wm * 32 + r + 8 * half;
        const int row1 = row0 + 16;
        Cr[(size_t)row0 * n + colg] = alpha * cr0[r] + ((row0 == colg) ? beta : 0.f);
        Ci[(size_t)row0 * n + colg] = alpha * ci0[r];
        Cr[(size_t)row1 * n + colg] = alpha * cr1[r] + ((row1 == colg) ? beta : 0.f);
        Ci[(size_t)row1 * n + colg] = alpha * ci1[r];
    }
}

// ---------------------------------------------------------------------------
// Small / elementwise kernels
// ---------------------------------------------------------------------------
__global__ void k_zero(float* p, int n) {
    int i = blockIdx.x * blockDim.x + threadIdx.x;
    if (i < n) p[i] = 0.f;
}

__global__ void k_deg(const int* __restrict__ ei, float* deg) {
    int e = blockIdx.x * blockDim.x + threadIdx.x;
    if (e < NEDG) atomicAdd(&deg[ei[NEDG + e]], 1.f);
}

__global__ void k_dinv(const float* deg, float* dinv) {
    int i = blockIdx.x * blockDim.x + threadIdx.x;
    if (i < NN) dinv[i] = rsqrtf(deg[i] + 1.f);  // +1 self-loop
}

__global__ void k_s1(const int* __restrict__ ei, const float* dinv, float* s1) {
    int e = blockIdx.x * blockDim.x + threadIdx.x;
    if (e < NEDG) {
        int s = ei[e], d = ei[NEDG + e];
        atomicAdd(&s1[d], dinv[s] * dinv[d]);
    }
}

__global__ void k_layer1(const float* s1, const float* dinv,
                         const float* Wg1, const float* bg1, float* x1) {
    int i = blockIdx.x * blockDim.x + threadIdx.x;  // NN*HD
    int nd = i >> 6, j = i & 63;
    float s = s1[nd] + dinv[nd] * dinv[nd];
    float v = s * Wg1[j] + bg1[j];
    x1[i] = v > 0.f ? v : 0.f;
}

__global__ void k_h2(const float* __restrict__ x1, const float* __restrict__ Wg2, float* h2) {
    int i = blockIdx.x * blockDim.x + threadIdx.x;  // NN*HD
    int nd = i >> 6, j = i & 63;
    float acc = 0.f;
    #pragma unroll 8
    for (int k = 0; k < HD; ++k) acc += x1[nd * HD + k] * Wg2[k * HD + j];
    h2[i] = acc;
}

__global__ void k_x2init(const float* h2, const float* dinv, const float* bg2, float* x2) {
    int i = blockIdx.x * blockDim.x + threadIdx.x;  // NN*HD
    int nd = i >> 6, j = i & 63;
    x2[i] = bg2[j] + dinv[nd] * dinv[nd] * h2[i];   // self-loop term
}

__global__ void k_x2edge(const int* __restrict__ ei, const float* dinv,
                         const float* h2, float* x2) {
    int i = blockIdx.x * blockDim.x + threadIdx.x;  // NEDG*HD
    int e = i >> 6, j = i & 63;
    int s = ei[e], d = ei[NEDG + e];
    atomicAdd(&x2[d * HD + j], dinv[s] * dinv[d] * h2[s * HD + j]);
}

__global__ void k_gsum(const float* x2, float* gsum) {
    int i = blockIdx.x * blockDim.x + threadIdx.x;  // NN*HD
    atomicAdd(&gsum[i & 63], x2[i]);
}

__global__ void k_mlp(const float* gsum, const float* Wp1, const float* bp1,
                      const float* Wp2, const float* bp2, float* m2) {
    __shared__ float g[HD];
    __shared__ float m1[HD];
    int t = threadIdx.x;
    g[t] = gsum[t] * (1.f / (float)NN);
    __syncthreads();
    float a = bp1[t];
    for (int k = 0; k < HD; ++k) a += g[k] * Wp1[k * HD + t];
    m1[t] = a > 0.f ? a : 0.f;
    __syncthreads();
    if (t < HD / 2) {
        float b = bp2[t];
        for (int k = 0; k < HD; ++k) b += m1[k] * Wp2[k * (HD / 2) + t];
        m2[t] = b > 0.f ? b : 0.f;
    }
}

__global__ void k_P(const float* __restrict__ m2g, const float* __restrict__ Wp3,
                    const float* __restrict__ bp3, float* P) {
    __shared__ float m2s[HD / 2];
    if (threadIdx.x < HD / 2) m2s[threadIdx.x] = m2g[threadIdx.x];
    __syncthreads();
    int m = blockIdx.x * blockDim.x + threadIdx.x;  // NN*NN
    float acc = bp3[m];
    #pragma unroll
    for (int k = 0; k < HD / 2; ++k) acc += m2s[k] * Wp3[(size_t)k * (NN * NN) + m];
    P[m] = tanhf(acc);
}

// B = A * 2^-8 where A = 0.5*0.05*((P-P^T) + i(P+P^T))
__global__ void k_buildB(const float* __restrict__ P, float* Bre, float* Bim) {
    int idx = blockIdx.x * blockDim.x + threadIdx.x;  // NN*NN
    int i = idx >> 10, j = idx & 1023;
    const float SC = 0.025f / 256.f;
    float pij = P[idx], pji = P[j * NN + i];
    Bre[idx] = (pij - pji) * SC;
    Bim[idx] = (pij + pji) * SC;
}

// D = a*S + b*I (complex)
__global__ void k_axpyI(const float* sr, const float* si, float* dr, float* di,
                        float a, float b) {
    int idx = blockIdx.x * blockDim.x + threadIdx.x;  // NN*NN
    int i = idx >> 10, j = idx & 1023;
    dr[idx] = a * sr[idx] + ((i == j) ? b : 0.f);
    di[idx] = a * si[idx];
}

// C <- (C + C^H)/2 + eps*I, in place
__global__ void k_symeps(float* Cr, float* Ci) {
    int idx = blockIdx.x * blockDim.x + threadIdx.x;  // NN*NN
    int i = idx >> 10, j = idx & 1023;
    if (j < i) return;
    float rr = 0.5f * (Cr[i * NN + j] + Cr[j * NN + i]);
    float im = 0.5f * (Ci[i * NN + j] - Ci[j * NN + i]);
    float d  = (i == j) ? 1e-8f : 0.f;
    Cr[i * NN + j] = rr + d; Cr[j * NN + i] = rr + d;
    Ci[i * NN + j] = im;     Ci[j * NN + i] = -im;
}

__global__ void k_fro(const float* __restrict__ r, const float* __restrict__ im, float* sc) {
    __shared__ float sh[256];
    float acc = 0.f;
    for (int i = blockIdx.x * blockDim.x + threadIdx.x; i < NN * NN;
         i += gridDim.x * blockDim.x) {
        float a = r[i], b = im[i];
        acc += a * a + b * b;
    }
    sh[threadIdx.x] = acc;
    __syncthreads();
    for (int s = 128; s > 0; s >>= 1) {
        if (threadIdx.x < s) sh[threadIdx.x] += sh[threadIdx.x + s];
        __syncthreads();
    }
    if (threadIdx.x == 0) atomicAdd(sc, sh[0]);
}

// Y = C/||C||_F, Z = I
__global__ void k_nsinit(const float* Cr, const float* Ci, const float* sc,
                         float* yr, float* yi, float* zr, float* zi) {
    int idx = blockIdx.x * blockDim.x + threadIdx.x;  // NN*NN
    int i = idx >> 10, j = idx & 1023;
    float inv = 1.f / (sqrtf(sc[0]) + 1e-30f);
    yr[idx] = Cr[idx] * inv;
    yi[idx] = Ci[idx] * inv;
    zr[idx] = (i == j) ? 1.f : 0.f;
    zi[idx] = 0.f;
}

// S = sqrt(||C||_F) * Y  (||C||_F = sqrt(sumsq) -> scale = sumsq^0.25)
__global__ void k_scaleY(const float* yr, const float* yi, const float* sc,
                         float* Sr, float* Si) {
    int idx = blockIdx.x * blockDim.x + threadIdx.x;
    float s = sqrtf(sqrtf(sc[0] + 1e-30f));
    Sr[idx] = s * yr[idx];
    Si[idx] = s * yi[idx];
}

// U_dilation = [[U, S1], [S2, -U^H]]  (complex64 interleaved re,im)
__global__ void k_out(const float* Ur, const float* Ui,
                      const float* S1r, const float* S1i,
                      const float* S2r, const float* S2i, float* out) {
    int idx = blockIdx.x * blockDim.x + threadIdx.x;  // (2N)^2
    int i = idx >> 11, j = idx & 2047;
    float re, im;
    if (i < NN) {
        if (j < NN) { re = Ur[i * NN + j];        im = Ui[i * NN + j]; }
        else        { re = S1r[i * NN + (j - NN)]; im = S1i[i * NN + (j - NN)]; }
    } else {
        int ii = i - NN;
        if (j < NN) { re = S2r[ii * NN + j]; im = S2i[ii * NN + j]; }
        else {  // -conj(U[j',i'])
            int jj = j - NN;
            re = -Ur[jj * NN + ii];
            im =  Ui[jj * NN + ii];
        }
    }
    out[2 * idx]     = re;
    out[2 * idx + 1] = im;
}

__global__ void k_copyU(const float* Ur, const float* Ui, float* out) {
    int m = blockIdx.x * blockDim.x + threadIdx.x;  // NN*NN
    out[2 * m]     = Ur[m];
    out[2 * m + 1] = Ui[m];
}

// ---------------------------------------------------------------------------
static void cgemm(const float* Ar, const float* Ai, const float* Br, const float* Bi,
                  float* Cr, float* Ci, int opA, int opB, float alpha, float beta,
                  hipStream_t s) {
    dim3 grid(NN / 64, NN / 64);
    k_cgemm<<<grid, 256, 0, s>>>(Ar, Ai, Br, Bi, Cr, Ci, NN, opA, opB, alpha, beta);
}

extern "C" void kernel_launch(void* const* d_in, const int* in_sizes, int n_in,
                              void* d_out, int out_size, void* d_ws, size_t ws_size,
                              hipStream_t stream) {
    (void)in_sizes; (void)n_in; (void)out_size; (void)ws_size;
    const int*   ei  = (const int*)  d_in[0];
    const float* Wg1 = (const float*)d_in[1];
    const float* bg1 = (const float*)d_in[2];
    const float* Wg2 = (const float*)d_in[3];
    const float* bg2 = (const float*)d_in[4];
    const float* Wp1 = (const float*)d_in[5];
    const float* bp1 = (const float*)d_in[6];
    const float* Wp2 = (const float*)d_in[7];
    const float* bp2 = (const float*)d_in[8];
    const float* Wp3 = (const float*)d_in[9];
    const float* bp3 = (const float*)d_in[10];
    float* out = (float*)d_out;

    const size_t PLSZ = (size_t)NN * NN;
    float* ws = (float*)d_ws;
    auto pl = [&](int i) { return ws + (size_t)i * PLSZ; };

    float* P   = pl(0);
    float* Bre = pl(1);  float* Bim = pl(2);
    float* X0r = pl(3);  float* X0i = pl(4);
    float* X1r = pl(5);  float* X1i = pl(6);
    float* C1r = pl(7);  float* C1i = pl(8);
    float* C2r = pl(9);  float* C2i = pl(10);
    float* S1r = pl(11); float* S1i = pl(12);
    float* S2r = pl(13); float* S2i = pl(14);
    // Newton-Schulz workspace: planes 15..22
    float* small = pl(23);
    float* deg  = small + 0;
    float* dinv = small + 1024;
    float* s1   = small + 2048;
    float* scal = small + 3072;   // 2 floats
    float* gsum = small + 4096;   // 64
    float* m2   = small + 4224;   // 32
    float* x1   = small + 8192;   // 65536
    float* h2   = small + 73728;  // 65536
    float* x2   = small + 139264; // 65536

    const int PG  = (NN * NN) / 256;  // 4096 blocks for full-plane elementwise

    // ---- GCN encoder + pooling + MLP head (memory-bound, scalar) ----
    k_zero<<<4, 256, 0, stream>>>(deg, NN);
    k_zero<<<4, 256, 0, stream>>>(s1, NN);
    k_zero<<<1, 64, 0, stream>>>(gsum, HD);
    k_zero<<<1, 64, 0, stream>>>(scal, 2);
    k_deg<<<NEDG / 256, 256, 0, stream>>>(ei, deg);
    k_dinv<<<NN / 256, 256, 0, stream>>>(deg, dinv);
    k_s1<<<NEDG / 256, 256, 0, stream>>>(ei, dinv, s1);
    k_layer1<<<(NN * HD) / 256, 256, 0, stream>>>(s1, dinv, Wg1, bg1, x1);
    k_h2<<<(NN * HD) / 256, 256, 0, stream>>>(x1, Wg2, h2);
    k_x2init<<<(NN * HD) / 256, 256, 0, stream>>>(h2, dinv, bg2, x2);
    k_x2edge<<<(NEDG * HD) / 256, 256, 0, stream>>>(ei, dinv, h2, x2);
    k_gsum<<<(NN * HD) / 256, 256, 0, stream>>>(x2, gsum);
    k_mlp<<<1, 64, 0, stream>>>(gsum, Wp1, bp1, Wp2, bp2, m2);
    k_P<<<PG, 256, 0, stream>>>(m2, Wp3, bp3, P);          // 134 MB stream of Wp3
    k_buildB<<<PG, 256, 0, stream>>>(P, Bre, Bim);         // B = A / 2^8

    // ---- expm(A): degree-8 Taylor (Horner) + 8 squarings, all WMMA GEMMs ----
    k_axpyI<<<PG, 256, 0, stream>>>(Bre, Bim, X0r, X0i, 1.f / 8.f, 1.f);
    float *xr = X0r, *xi = X0i, *yr = X1r, *yi = X1i;
    for (int k = 7; k >= 1; --k) {  // X <- I + (B/k) * X
        cgemm(Bre, Bim, xr, xi, yr, yi, 0, 0, 1.f / (float)k, 1.f, stream);
        float* t;
        t = xr; xr = yr; yr = t;  t = xi; xi = yi; yi = t;
    }
    for (int t8 = 0; t8 < 8; ++t8) {  // X <- X * X
        cgemm(xr, xi, xr, xi, yr, yi, 0, 0, 1.f, 0.f, stream);
        float* t;
        t = xr; xr = yr; yr = t;  t = xi; xi = yi; yi = t;
    }
    float* Ur = xr; float* Ui = xi;

    // ---- defect Gram matrices ----
    cgemm(Ur, Ui, Ur, Ui, C1r, C1i, 1, 0, -1.f, 1.f, stream);  // I - U^H U
    cgemm(Ur, Ui, Ur, Ui, C2r, C2i, 0, 1, -1.f, 1.f, stream);  // I - U U^H
    k_symeps<<<PG, 256, 0, stream>>>(C1r, C1i);
    k_symeps<<<PG, 256, 0, stream>>>(C2r, C2i);

    // ---- Hermitian-PSD sqrt via coupled Newton-Schulz (GEMM-only) ----
    auto nssqrt = [&](float* Cr, float* Ci, float* Sr, float* Si, float* sc) {
        float *nyr = pl(15), *nyi = pl(16);
        float *nzr = pl(17), *nzi = pl(18);
        float *nwr = pl(19), *nwi = pl(20);
        float *ntr = pl(21), *nti = pl(22);
        k_fro<<<256, 256, 0, stream>>>(Cr, Ci, sc);
        k_nsinit<<<PG, 256, 0, stream>>>(Cr, Ci, sc, nyr, nyi, nzr, nzi);
        for (int it = 0; it < 10; ++it) {
            cgemm(nzr, nzi, nyr, nyi, nwr, nwi, 0, 0, -0.5f, 1.5f, stream); // W=1.5I-0.5ZY
            cgemm(nyr, nyi, nwr, nwi, ntr, nti, 0, 0, 1.f, 0.f, stream);    // T=Y*W
            cgemm(nwr, nwi, nzr, nzi, nyr, nyi, 0, 0, 1.f, 0.f, stream);    // oldY<-W*Z
            float *oyr = nyr, *oyi = nyi;
            nyr = ntr; nyi = nti;      // Y := T
            ntr = nzr; nti = nzi;      // next T target := old Z
            nzr = oyr; nzi = oyi;      // Z := W*Z (in old Y planes)
        }
        k_scaleY<<<PG, 256, 0, stream>>>(nyr, nyi, sc, Sr, Si);
    };
    nssqrt(C1r, C1i, S1r, S1i, scal + 0);
    nssqrt(C2r, C2i, S2r, S2i, scal + 1);

    // ---- assemble outputs: U_dilation (2N x 2N) then U (N x N), complex64 ----
    k_out<<<(4 * NN * NN) / 256, 256, 0, stream>>>(Ur, Ui, S1r, S1i, S2r, S2i, out);
    k_copyU<<<PG, 256, 0, stream>>>(Ur, Ui, out + (size_t)2 * (2 * NN) * (2 * NN));
}